// GAE_10411000726026
// MI455X (gfx1250) — compile-verified
//
#include <hip/hip_runtime.h>

// Problem sizes (fixed by the reference)
#define N_NODES 10000
#define N_EDGES 320000
#define F_IN    512
#define NHID    32
#define LATENT  16
#define NT      625          // N_NODES / 16 output tiles per dim (exact)

typedef __attribute__((ext_vector_type(2))) float v2f;
typedef __attribute__((ext_vector_type(8))) float v8f;

// ---------------------------------------------------------------------------
// zero a float range (graph-replay-safe init of accumulation buffers)
// ---------------------------------------------------------------------------
__global__ __launch_bounds__(256) void gae_zero(float* __restrict__ p, int n) {
    int i = blockIdx.x * 256 + threadIdx.x;
    if (i < n) p[i] = 0.0f;
}

// ---------------------------------------------------------------------------
// h0 = x @ W1   [10000,512] x [512,32]  — one thread per output element.
// x-row reads broadcast across the 32 lanes of a wave; W1 (64KB) stays hot
// in L1/L2. Compute is 0.33 GFLOP: negligible vs the decoder kernel.
// ---------------------------------------------------------------------------
__global__ __launch_bounds__(256) void gae_gemm1(const float* __restrict__ x,
                                                 const float* __restrict__ W1,
                                                 float* __restrict__ h0) {
    int t = blockIdx.x * 256 + threadIdx.x;
    if (t >= N_NODES * NHID) return;
    int row = t >> 5;
    int col = t & 31;
    const float* xr = x + (size_t)row * F_IN;
    float acc = 0.0f;
#pragma unroll 8
    for (int k = 0; k < F_IN; ++k)
        acc = fmaf(xr[k], W1[k * NHID + col], acc);
    h0[t] = acc;
}

// ---------------------------------------------------------------------------
// spmm over 32 features: one wave per edge, lane = feature.
// Coalesced gather of h_in[src][0..31] and coalesced f32 atomics to h_out[dst].
// ---------------------------------------------------------------------------
__global__ __launch_bounds__(256) void gae_spmm32(const int* __restrict__ src,
                                                  const int* __restrict__ dst,
                                                  const float* __restrict__ w,
                                                  const float* __restrict__ hin,
                                                  float* __restrict__ hout) {
    int e    = (blockIdx.x * 256 + threadIdx.x) >> 5;
    int lane = threadIdx.x & 31;
    if (e >= N_EDGES) return;
    int   s  = src[e];
    int   d  = dst[e];
    float wt = w[e];
    float v  = wt * hin[(size_t)s * NHID + lane];
    atomicAdd(&hout[(size_t)d * NHID + lane], v);
}

// ---------------------------------------------------------------------------
// z0 = h @ W2   [10000,32] x [32,16] — one thread per output element.
// ---------------------------------------------------------------------------
__global__ __launch_bounds__(256) void gae_gemm2(const float* __restrict__ h,
                                                 const float* __restrict__ W2,
                                                 float* __restrict__ z0) {
    int t = blockIdx.x * 256 + threadIdx.x;
    if (t >= N_NODES * LATENT) return;
    int row = t >> 4;
    int col = t & 15;
    const float* hr = h + (size_t)row * NHID;
    float acc = 0.0f;
#pragma unroll
    for (int k = 0; k < NHID; ++k)
        acc = fmaf(hr[k], W2[k * LATENT + col], acc);
    z0[t] = acc;
}

// ---------------------------------------------------------------------------
// spmm over 16 features: half-wave per edge (lane/16 = edge slot).
// ---------------------------------------------------------------------------
__global__ __launch_bounds__(256) void gae_spmm16(const int* __restrict__ src,
                                                  const int* __restrict__ dst,
                                                  const float* __restrict__ w,
                                                  const float* __restrict__ zin,
                                                  float* __restrict__ zout) {
    int gt = blockIdx.x * 256 + threadIdx.x;
    int e  = gt >> 4;
    int f  = gt & 15;
    if (e >= N_EDGES) return;
    int   s  = src[e];
    int   d  = dst[e];
    float wt = w[e];
    float v  = wt * zin[(size_t)s * LATENT + f];
    atomicAdd(&zout[(size_t)d * LATENT + f], v);
}

// ---------------------------------------------------------------------------
// copy z into the second output slot (d_out + N*N)
// ---------------------------------------------------------------------------
__global__ __launch_bounds__(256) void gae_copyz(const float* __restrict__ z,
                                                 float* __restrict__ out) {
    int i = blockIdx.x * 256 + threadIdx.x;
    if (i < N_NODES * LATENT) out[i] = z[i];
}

// ---------------------------------------------------------------------------
// Decoder: adj = sigmoid(z @ z^T), N=10000, K=16, via V_WMMA_F32_16X16X4_F32.
// One wave computes one 16x16 tile with 4 chained K=4 f32 WMMAs.
// 8 waves/block cover 8 adjacent column tiles of one row tile so the 400MB
// store stream is contiguous per row across the block. z (640KB) stays in L2,
// so this kernel runs at the HBM store roofline (~400MB / 23.3TB/s ~= 17us).
//
// Sigmoid epilogue uses the hardware reciprocal (v_rcp_f32) instead of an
// IEEE divide: the div_scale/div_fmas Newton sequence (~12 VALU ops/elem seen
// in round-1 disasm) would cost ~60us of VALU across 1e8 outputs — 3x the
// store roofline. rcp + v_exp_f32 keeps the epilogue ~4 ops/elem (~bandwidth
// bound again). Stores are non-temporal: 400MB write-once stream should not
// rinse through L2.
//
// f32 16x4 A layout (ISA 7.12.2): VGPR0 = {K0 | K2}, VGPR1 = {K1 | K3} over
// lane halves -> per lane one float2 at k0 = kk*4 + (lane>=16 ? 2 : 0).
// B (4x16) mirrored: lane%16 = N column (= z row of the column tile).
// D layout: VGPR r holds row M = r (lanes 0-15) / r+8 (lanes 16-31).
// ---------------------------------------------------------------------------
__global__ __launch_bounds__(256) void gae_decoder(const float* __restrict__ z,
                                                   float* __restrict__ out) {
    const int lane = threadIdx.x & 31;
    const int wave = threadIdx.x >> 5;           // 0..7
    const int bm   = blockIdx.y;                 // 0..624 row tile
    const int bn   = blockIdx.x * 8 + wave;      // column tile
    if (bn >= NT) return;                        // whole-wave exit: EXEC stays all-1s

    const int half  = lane >> 4;                 // 0 | 1
    const int idx16 = lane & 15;
    const int arow  = bm * 16 + idx16;           // A-matrix row (M)
    const int brow  = bn * 16 + idx16;           // B-matrix column (N) = z row

    const v2f* z2 = reinterpret_cast<const v2f*>(z);   // z rows are 16 floats

    // Preload all 4 K-chunks (each one global_load_b64 per operand).
    v2f a0 = z2[(size_t)arow * 8 + 0 * 2 + half];
    v2f a1 = z2[(size_t)arow * 8 + 1 * 2 + half];
    v2f a2 = z2[(size_t)arow * 8 + 2 * 2 + half];
    v2f a3 = z2[(size_t)arow * 8 + 3 * 2 + half];
    v2f b0 = z2[(size_t)brow * 8 + 0 * 2 + half];
    v2f b1 = z2[(size_t)brow * 8 + 1 * 2 + half];
    v2f b2 = z2[(size_t)brow * 8 + 2 * 2 + half];
    v2f b3 = z2[(size_t)brow * 8 + 3 * 2 + half];

    v8f c = {0.f, 0.f, 0.f, 0.f, 0.f, 0.f, 0.f, 0.f};
    c = __builtin_amdgcn_wmma_f32_16x16x4_f32(false, a0, false, b0, (short)0, c, false, false);
    c = __builtin_amdgcn_wmma_f32_16x16x4_f32(false, a1, false, b1, (short)0, c, false, false);
    c = __builtin_amdgcn_wmma_f32_16x16x4_f32(false, a2, false, b2, (short)0, c, false, false);
    c = __builtin_amdgcn_wmma_f32_16x16x4_f32(false, a3, false, b3, (short)0, c, false, false);

    const int col = bn * 16 + idx16;
#pragma unroll
    for (int r = 0; r < 8; ++r) {
        int   row = bm * 16 + r + half * 8;
        // sigmoid(v) = rcp(1 + exp(-v)); v_rcp_f32 (~1 ulp) instead of IEEE div
        float ex  = __expf(-c[r]);
        float sig = __builtin_amdgcn_rcpf(1.0f + ex);
        __builtin_nontemporal_store(sig, &out[(size_t)row * N_NODES + col]);
    }
}

// ---------------------------------------------------------------------------
// Host-side orchestration (all launches on `stream`, graph-capture safe)
// ---------------------------------------------------------------------------
extern "C" void kernel_launch(void* const* d_in, const int* in_sizes, int n_in,
                              void* d_out, int out_size, void* d_ws, size_t ws_size,
                              hipStream_t stream) {
    const float* x  = (const float*)d_in[0];                // [10000,512]
    const float* W1 = (const float*)d_in[1];                // [512,32]
    const float* W2 = (const float*)d_in[2];                // [32,16]
    const int*   ei = (const int*)d_in[3];                  // [2,320000]
    const float* ew = (const float*)d_in[4];                // [320000]

    const int* src = ei;                                    // edge_index[0]
    const int* dst = ei + N_EDGES;                          // edge_index[1]

    float* ws = (float*)d_ws;
    float* h0 = ws;                                         // 320000 floats
    float* h  = ws + (size_t)N_NODES * NHID;                // 320000 floats
    float* z0 = h + (size_t)N_NODES * NHID;                 // 160000 floats
    float* z  = z0 + (size_t)N_NODES * LATENT;              // 160000 floats

    float* adj  = (float*)d_out;                            // [10000,10000]
    float* zout = adj + (size_t)N_NODES * N_NODES;          // [10000,16]

    // 1) zero accumulation buffers (h, z)
    {
        int n = N_NODES * NHID;
        gae_zero<<<(n + 255) / 256, 256, 0, stream>>>(h, n);
        int m = N_NODES * LATENT;
        gae_zero<<<(m + 255) / 256, 256, 0, stream>>>(z, m);
    }
    // 2) h0 = x @ W1
    {
        int n = N_NODES * NHID;
        gae_gemm1<<<(n + 255) / 256, 256, 0, stream>>>(x, W1, h0);
    }
    // 3) h = spmm(h0)
    gae_spmm32<<<(N_EDGES * 32 + 255) / 256, 256, 0, stream>>>(src, dst, ew, h0, h);
    // 4) z0 = h @ W2
    {
        int n = N_NODES * LATENT;
        gae_gemm2<<<(n + 255) / 256, 256, 0, stream>>>(h, W2, z0);
    }
    // 5) z = spmm(z0)
    gae_spmm16<<<(N_EDGES * 16 + 255) / 256, 256, 0, stream>>>(src, dst, ew, z0, z);
    // 6) second output: z
    gae_copyz<<<(N_NODES * LATENT + 255) / 256, 256, 0, stream>>>(z, zout);
    // 7) adj = sigmoid(z @ z^T) via f32 WMMA
    {
        dim3 grid((NT + 7) / 8, NT);   // 79 x 625 blocks, 8 waves/block
        gae_decoder<<<grid, 256, 0, stream>>>(z, adj);
    }
}